// CorrLayerDownsample_63943473103015
// MI455X (gfx1250) — compile-verified
//
#include <hip/hip_runtime.h>
#include <hip/hip_bf16.h>

// ---------------------------------------------------------------------------
// CorrLayerDownsample for MI455X (gfx1250, wave32).
// Output [2, 384, 7]: 6 scale-pairs x 64 (l1,l2) channel pairs x 7 shifts.
// Shifts in sorted-union order: (0,0),(0,1),(0,2),(1,0),(1,1),(2,0),(-1,1).
//   same-scale pairs  -> exact spatial GEMM via V_WMMA_F32_16X16X4_F32
//   cross-scale pairs -> FFT2 to ws, then phased frequency reduction over the
//                        coarse spectrum support (factor 1/(h2*w2))
// All sizes are compile-time powers of two; the WMMA inner loop is fully
// branchless (zero-row trick + circularly extended LDS rows).
// ---------------------------------------------------------------------------

typedef __attribute__((ext_vector_type(2))) float v2f;
typedef __attribute__((ext_vector_type(8))) float v8f;

__device__ __constant__ int c_sdx[7] = {0, 0, 0, 1, 1, 2, -1};
__device__ __constant__ int c_sdy[7] = {0, 1, 2, 0, 1, 0, 1};

#define TWO_PI_F 6.28318530717958647692f

// ---------------------------------------------------------------------------
// Same-scale kernel: C[(s,l1),(l2)] = sum_{x,y} X[l1,(x+dx)%H,(y+dy)%W] * X[l2,x,y]
// GEMM: M=56 (pad 64 -> 4 wave tiles of 16), N=8 (pad 16), K = H*W row-by-row.
// grid.x = NB * (H / ROWS), blockDim = 128 (4 waves).
// LDS: 32 rows (4 dx-slots x 8 ch) of W+4 floats (circular ext) + 1 zero row.
// ---------------------------------------------------------------------------
template <int LOGW, int ROWS>
__global__ void __launch_bounds__(128)
same_scale_corr(const float* __restrict__ xin, int pairBase,
                float* __restrict__ out) {
  constexpr int W = 1 << LOGW;
  constexpr int H = W;
  constexpr int WS = W + 4;        // row stride (floats); multiple of 4
  constexpr int ZROW = 32 * WS;    // zero-filled row offset
  constexpr int SLABS = H / ROWS;  // power of two for all instantiations
  __shared__ float lds[33 * WS];

  const int b = blockIdx.x / SLABS;
  const int slab = blockIdx.x & (SLABS - 1);

  const int lane = threadIdx.x & 31;
  const int wave = threadIdx.x >> 5;
  const int m0 = wave * 16;

  // per-lane constant A-fragment metadata (rows of the (shift,l1) matrix)
  const int mrow = m0 + (lane & 15);
  const bool aval = (mrow < 56);
  const int sidx = aval ? (mrow >> 3) : 0;
  const int l1 = mrow & 7;
  const int slot = c_sdx[sidx] + 1;  // row slot for dx in {-1,0,1,2}
  const int dy = aval ? c_sdy[sidx] : 0;
  const int aIdx = (aval ? (slot * 8 + l1) * WS : ZROW) + dy;
  // B-fragment metadata (columns = l2); slot 1 == dx 0
  const int ncol = lane & 15;
  const int bBase = (ncol < 8) ? (8 + ncol) * WS : ZROW;
  const int khi = (lane >> 4) << 1;  // lanes 0-15: K+0/K+1, lanes 16-31: K+2/K+3

  // zero row init (synced by the staging barrier below)
  for (int idx = threadIdx.x; idx < WS; idx += 128) lds[ZROW + idx] = 0.0f;

  v8f acc = {};

  for (int xi = 0; xi < ROWS; ++xi) {
    const int xr = slab * ROWS + xi;

    // stage rows (xr-1 .. xr+2) mod H for all 8 channels into LDS, 16B chunks
    {
      constexpr int NV4 = 8 * W;       // float4 elements over 32 rows
      constexpr int PER_CH4 = W / 4;   // float4 per row
      for (int idx = threadIdx.x; idx < NV4; idx += 128) {
        const int row = idx >> (LOGW - 2);      // 0..31 = slot*8+ch
        const int y4 = idx & (PER_CH4 - 1);
        const int rs = row >> 3;
        const int ch = row & 7;
        int xs = xr + rs - 1;
        xs += (xs < 0) ? H : 0;
        xs -= (xs >= H) ? H : 0;
        *(float4*)&lds[row * WS + (y4 << 2)] =
            *(const float4*)&xin[((((size_t)b * 8 + ch) << LOGW) + xs) * W +
                                 (y4 << 2)];
      }
      // circular 4-float extension per row: lds[row][W+e] = X[row][e]
      const int row = threadIdx.x >> 2;  // 0..31
      const int e = threadIdx.x & 3;
      const int rs = row >> 3;
      const int ch = row & 7;
      int xs = xr + rs - 1;
      xs += (xs < 0) ? H : 0;
      xs -= (xs >= H) ? H : 0;
      lds[row * WS + W + e] =
          xin[((((size_t)b * 8 + ch) << LOGW) + xs) * W + e];
    }
    // prefetch a future row slab into cache (global_prefetch_b8)
    if (threadIdx.x < 8) {
      int xn = xr + 4;
      xn -= (xn >= H) ? H : 0;
      __builtin_prefetch(
          &xin[((((size_t)b * 8 + threadIdx.x) << LOGW) + xn) * W], 0, 1);
    }
    __syncthreads();

    // K loop over this row (K = W), 4 per WMMA step; fully branchless,
    // EXEC all-ones at every WMMA.
#pragma unroll 8
    for (int k = 0; k < W; k += 4) {
      const int kb = k + khi;
      v2f av, bv;
      av[0] = lds[aIdx + kb];                  // ds_load_b32
      av[1] = lds[aIdx + kb + 1];              // ds_load_b32
      bv = *(const v2f*)&lds[bBase + kb];      // ds_load_b64 (8B aligned)
      acc = __builtin_amdgcn_wmma_f32_16x16x4_f32(
          /*neg_a=*/false, av, /*neg_b=*/false, bv,
          /*c_mod=*/(short)0, acc, /*reuse_a=*/false, /*reuse_b=*/false);
    }
    __syncthreads();
  }

  // scatter C tile: VGPR i holds rows m0+i (lanes 0-15) / m0+8+i (lanes 16-31)
  const int rbase = m0 + ((lane >> 4) << 3);
  const int col = lane & 15;
  if (col < 8) {
#pragma unroll
    for (int i = 0; i < 8; ++i) {
      const int R = rbase + i;
      if (R < 56) {
        const int s = R >> 3;
        const int l1o = R & 7;
        atomicAdd(&out[(((size_t)b * 384) + pairBase + l1o * 8 + col) * 7 + s],
                  acc[i]);
      }
    }
  }
}

// ---------------------------------------------------------------------------
// Radix-2 DIT FFT along rows: one block per (image,row), blockDim = N/2.
// Fully unrolled stages (LOGN is a template parameter).
// ---------------------------------------------------------------------------
template <int LOGN>
__global__ void fft_rows(const float* __restrict__ xin,
                         float2* __restrict__ spec) {
  constexpr int N = 1 << LOGN;
  __shared__ float2 buf[N];
  const int row = blockIdx.x & (N - 1);  // h == w == N
  const int img = blockIdx.x >> LOGN;
  const float* src = xin + (((size_t)img << LOGN) + row) * N;
  const int t = threadIdx.x;  // N/2 threads

  buf[__brev((unsigned)t) >> (32 - LOGN)] = make_float2(src[t], 0.0f);
  buf[__brev((unsigned)(t + N / 2)) >> (32 - LOGN)] =
      make_float2(src[t + N / 2], 0.0f);
  __syncthreads();
#pragma unroll
  for (int st = 1; st <= LOGN; ++st) {
    const int len = 1 << st;
    const int half = len >> 1;
    const int j = t & (half - 1);
    const int base = (t >> (st - 1)) << st;
    float sn, cs;
    __sincosf(-TWO_PI_F * (float)j / (float)len, &sn, &cs);
    const float2 u = buf[base + j];
    const float2 v = buf[base + j + half];
    const float2 tv = make_float2(v.x * cs - v.y * sn, v.x * sn + v.y * cs);
    buf[base + j] = make_float2(u.x + tv.x, u.y + tv.y);
    buf[base + j + half] = make_float2(u.x - tv.x, u.y - tv.y);
    __syncthreads();
  }
  float2* dst = spec + (((size_t)img << LOGN) + row) * N;
  dst[t] = buf[t];
  dst[t + N / 2] = buf[t + N / 2];
}

// ---------------------------------------------------------------------------
// Radix-2 DIT FFT along columns (in place on spectra): block per (image,col).
// ---------------------------------------------------------------------------
template <int LOGN>
__global__ void fft_cols(float2* __restrict__ spec) {
  constexpr int N = 1 << LOGN;
  __shared__ float2 buf[N];
  const int col = blockIdx.x & (N - 1);
  const int img = blockIdx.x >> LOGN;
  float2* base0 = spec + ((size_t)img << (2 * LOGN)) + col;
  const int t = threadIdx.x;  // N/2 threads

  buf[__brev((unsigned)t) >> (32 - LOGN)] = base0[(size_t)t * N];
  buf[__brev((unsigned)(t + N / 2)) >> (32 - LOGN)] =
      base0[(size_t)(t + N / 2) * N];
  __syncthreads();
#pragma unroll
  for (int st = 1; st <= LOGN; ++st) {
    const int len = 1 << st;
    const int half = len >> 1;
    const int j = t & (half - 1);
    const int base = (t >> (st - 1)) << st;
    float sn, cs;
    __sincosf(-TWO_PI_F * (float)j / (float)len, &sn, &cs);
    const float2 u = buf[base + j];
    const float2 v = buf[base + j + half];
    const float2 tv = make_float2(v.x * cs - v.y * sn, v.x * sn + v.y * cs);
    buf[base + j] = make_float2(u.x + tv.x, u.y + tv.y);
    buf[base + j + half] = make_float2(u.x - tv.x, u.y - tv.y);
    __syncthreads();
  }
  base0[(size_t)t * N] = buf[t];
  base0[(size_t)(t + N / 2) * N] = buf[t + N / 2];
}

// ---------------------------------------------------------------------------
// Cross-scale: corr[d] = 1/(N2*N2) * Re( sum_{f in [-N2/2,N2/2)^2}
//              F1[f mod N1] * conj(F2[f mod N2]) * e^{+2pi i f.d/N1} )
// One block (256 thr) per (b, l1, l2); separable phase factors per bin.
// ---------------------------------------------------------------------------
template <int LOG1, int LOG2>
__global__ void __launch_bounds__(256)
cross_corr(const float2* __restrict__ s1, const float2* __restrict__ s2,
           int pairBase, float* __restrict__ out) {
  constexpr int N1 = 1 << LOG1;
  constexpr int N2 = 1 << LOG2;
  __shared__ float red[256];
  const int b = blockIdx.x >> 6;
  const int p = blockIdx.x & 63;
  const int l1 = p >> 3, l2 = p & 7;
  const float2* F1 = s1 + (((size_t)b * 8 + l1) << (2 * LOG1));
  const float2* F2 = s2 + (((size_t)b * 8 + l2) << (2 * LOG2));

  float a0 = 0.f, a1 = 0.f, a2 = 0.f, a3 = 0.f, a4 = 0.f, a5 = 0.f, a6 = 0.f;
  constexpr int BINS = N2 * N2;
  for (int tbin = threadIdx.x; tbin < BINS; tbin += blockDim.x) {
    const int ix2 = tbin >> LOG2;
    const int iy2 = tbin & (N2 - 1);
    const int fx = (ix2 < (N2 >> 1)) ? ix2 : ix2 - N2;
    const int fy = (iy2 < (N2 >> 1)) ? iy2 : iy2 - N2;
    const float2 A =
        F1[(((fx + N1) & (N1 - 1)) << LOG1) + ((fy + N1) & (N1 - 1))];
    const float2 B = F2[tbin];
    const float zr = A.x * B.x + A.y * B.y;  // Z = F1 * conj(F2)
    const float zi = A.y * B.x - A.x * B.y;
    float sx, cx, sy, cy;
    __sincosf(TWO_PI_F * (float)fx / (float)N1, &sx, &cx);
    __sincosf(TWO_PI_F * (float)fy / (float)N1, &sy, &cy);
    const float c2x = cx * cx - sx * sx, s2x = 2.f * cx * sx;
    const float c2y = cy * cy - sy * sy, s2y = 2.f * cy * sy;
    a0 += zr;                   // (0,0)
    a1 += zr * cy - zi * sy;    // (0,1)
    a2 += zr * c2y - zi * s2y;  // (0,2)
    a3 += zr * cx - zi * sx;    // (1,0)
    const float xr = zr * cx - zi * sx, xi = zr * sx + zi * cx;  // Z*e^{i tx}
    a4 += xr * cy - xi * sy;    // (1,1)
    a5 += zr * c2x - zi * s2x;  // (2,0)
    const float mr = zr * cx + zi * sx, mi = zi * cx - zr * sx;  // Z*e^{-i tx}
    a6 += mr * cy - mi * sy;    // (-1,1)
  }

  const float inv = 1.0f / (float)BINS;
  float accs[7] = {a0, a1, a2, a3, a4, a5, a6};
#pragma unroll
  for (int s = 0; s < 7; ++s) {
    red[threadIdx.x] = accs[s];
    __syncthreads();
    for (int off = 128; off > 0; off >>= 1) {
      if (threadIdx.x < off) red[threadIdx.x] += red[threadIdx.x + off];
      __syncthreads();
    }
    if (threadIdx.x == 0)
      out[(((size_t)b * 384) + pairBase + p) * 7 + s] = red[0] * inv;
    __syncthreads();
  }
}

// ---------------------------------------------------------------------------
extern "C" void kernel_launch(void* const* d_in, const int* in_sizes, int n_in,
                              void* d_out, int out_size, void* d_ws,
                              size_t ws_size, hipStream_t stream) {
  const float* x0 = (const float*)d_in[0];  // [2,8,256,256]
  const float* x1 = (const float*)d_in[1];  // [2,8,128,128]
  const float* x2 = (const float*)d_in[2];  // [2,8,64,64]
  float* out = (float*)d_out;               // [2,384,7]

  hipMemsetAsync(d_out, 0, (size_t)out_size * sizeof(float), stream);

  // same-scale pairs (pair indices 0,3,5 -> bases 0,192,320), WMMA path
  same_scale_corr<8, 2><<<dim3(2 * 128), 128, 0, stream>>>(x0, 0, out);
  same_scale_corr<7, 1><<<dim3(2 * 128), 128, 0, stream>>>(x1, 192, out);
  same_scale_corr<6, 1><<<dim3(2 * 64), 128, 0, stream>>>(x2, 320, out);

  // spectra in workspace: 16 images per scale, complex64 (~10.5 MB)
  const size_t n0 = (size_t)16 * 256 * 256;
  const size_t n1 = (size_t)16 * 128 * 128;
  const size_t n2 = (size_t)16 * 64 * 64;
  if (ws_size >= (n0 + n1 + n2) * sizeof(float2)) {
    float2* spec0 = (float2*)d_ws;
    float2* spec1 = spec0 + n0;
    float2* spec2 = spec1 + n1;

    fft_rows<8><<<dim3(16 * 256), 128, 0, stream>>>(x0, spec0);
    fft_cols<8><<<dim3(16 * 256), 128, 0, stream>>>(spec0);
    fft_rows<7><<<dim3(16 * 128), 64, 0, stream>>>(x1, spec1);
    fft_cols<7><<<dim3(16 * 128), 64, 0, stream>>>(spec1);
    fft_rows<6><<<dim3(16 * 64), 32, 0, stream>>>(x2, spec2);
    fft_cols<6><<<dim3(16 * 64), 32, 0, stream>>>(spec2);

    // cross-scale pairs: (0,1)->base 64, (0,2)->128, (1,2)->256
    cross_corr<8, 7><<<dim3(128), 256, 0, stream>>>(spec0, spec1, 64, out);
    cross_corr<8, 6><<<dim3(128), 256, 0, stream>>>(spec0, spec2, 128, out);
    cross_corr<7, 6><<<dim3(128), 256, 0, stream>>>(spec1, spec2, 256, out);
  }
}